// FFM_Layer_19387482374159
// MI455X (gfx1250) — compile-verified
//
#include <hip/hip_runtime.h>

// FFM layer for MI455X (gfx1250, wave32).
//
// Roofline: 256 MB of gathered v-rows at 23.3 TB/s ~= 11 us; compute is ~80 MFLOP
// (noise). Strategy: one wave per batch element, 16 independent global_load_b128
// per lane (one 512B v-row per wave-wide load), accumulate lv in VGPRs, then
// compute the pairwise-interaction term as the off-diagonal sum of the Gram
// matrix lv*lv^T with two chained V_WMMA_F32_16X16X4_F32 (K=0..3 and K=4..7),
// exploiting that for the documented 32-bit A layout the same VGPR pair serves
// as both A (lv) and B (lv^T) of the symmetric product.

typedef __attribute__((ext_vector_type(2))) float v2f;
typedef __attribute__((ext_vector_type(8))) float v8f;

#define NFIELD 16
#define NK 8
#define PER_FIELD 65536

__global__ __launch_bounds__(256) void ffm_kernel(
    const int* __restrict__ inputs,   // [B, 16] raw per-field ids
    const float* __restrict__ w0,     // [1]
    const float* __restrict__ w,      // [FEATURE_LEN]
    const float* __restrict__ v,      // [FEATURE_LEN, 16, 8] row-major (128 floats/row)
    float* __restrict__ out,          // [B]
    int B)
{
    const int lane = threadIdx.x & 31;
    const int wid  = (int)(blockIdx.x * (blockDim.x >> 5)) + (threadIdx.x >> 5);
    if (wid >= B) return;  // wave-uniform (wid identical across the wave)

    // ---- index gather + first-order weights: lanes 0..15 own one field each
    int   row   = 0;
    float wpart = 0.0f;
    if (lane < NFIELD) {
        int id = inputs[wid * NFIELD + lane];
        row    = id + lane * PER_FIELD;          // per-field vocab offset
        wpart  = w[row];                         // linear term gather
    }

    // ---- accumulate lv = sum_g v[row_g] in natural layout:
    // lane L holds row elements 4L..4L+3  (f = L/2, k = (L&1)*4 + 0..3)
    float ax = 0.f, ay = 0.f, az = 0.f, aw = 0.f;
    const float4* v4 = reinterpret_cast<const float4*>(v);
#pragma unroll
    for (int g = 0; g < NFIELD; ++g) {
        int rowg = __shfl(row, g, 32);           // broadcast field g's row id
        float4 t = v4[(size_t)rowg * 32 + lane]; // global_load_b128, coalesced 512B/row
        ax += t.x; ay += t.y; az += t.z; aw += t.w;
    }

    // ---- rearrange into WMMA 32-bit A-matrix layout (16x4 per chunk):
    // chunk0 (k=0..3): VGPR0 = lv[f][0] (lanes<16) / lv[f][2] (lanes>=16), VGPR1 = lv[f][1]/lv[f][3]
    // chunk1 (k=4..7): same with k+4. Source: lane 2f holds k=0..3, lane 2f+1 holds k=4..7.
    const int s0 = 2 * (lane & 15);
    const int s1 = s0 + 1;
    float x0 = __shfl(ax, s0, 32), y0 = __shfl(ay, s0, 32);
    float z0 = __shfl(az, s0, 32), w0v = __shfl(aw, s0, 32);
    float x1 = __shfl(ax, s1, 32), y1 = __shfl(ay, s1, 32);
    float z1 = __shfl(az, s1, 32), w1v = __shfl(aw, s1, 32);
    const bool lo = (lane < 16);
    v2f a0, a1;
    a0.x = lo ? x0 : z0;  a0.y = lo ? y0 : w0v;   // K-chunk 0..3
    a1.x = lo ? x1 : z1;  a1.y = lo ? y1 : w1v;   // K-chunk 4..7

    // ---- Gram matrix G = lv * lv^T via two chained f32 WMMAs (same regs as A and B)
    v8f c = {0.f, 0.f, 0.f, 0.f, 0.f, 0.f, 0.f, 0.f};
    c = __builtin_amdgcn_wmma_f32_16x16x4_f32(false, a0, false, a0, (short)0, c, false, false);
    c = __builtin_amdgcn_wmma_f32_16x16x4_f32(false, a1, false, a1, (short)0, c, false, false);

    // ---- second order = 0.5 * (sum_all(G) - trace(G))
    // C/D layout: VGPR r: lanes 0-15 -> G[M=r][N=lane]; lanes 16-31 -> G[M=r+8][N=lane-16]
    float sall = 0.f, diag = 0.f;
#pragma unroll
    for (int r = 0; r < 8; ++r) {
        float e = c[r];
        sall += e;
        // diagonal: lanes 0..7 at r==lane, lanes 24..31 at r==lane-24
        diag += ((lane == r) || (lane == r + 24)) ? e : 0.f;
    }

    // ---- wave32 butterfly reductions
#pragma unroll
    for (int m = 16; m >= 1; m >>= 1) {
        sall  += __shfl_xor(sall,  m, 32);
        diag  += __shfl_xor(diag,  m, 32);
        wpart += __shfl_xor(wpart, m, 32);
    }

    if (lane == 0) {
        out[wid] = w0[0] + wpart + 0.5f * (sall - diag);
    }
}

extern "C" void kernel_launch(void* const* d_in, const int* in_sizes, int n_in,
                              void* d_out, int out_size, void* d_ws, size_t ws_size,
                              hipStream_t stream) {
    const int*   inputs = (const int*)  d_in[0];
    const float* w0     = (const float*)d_in[1];
    const float* w      = (const float*)d_in[2];
    const float* v      = (const float*)d_in[3];
    float*       out    = (float*)d_out;

    const int B = in_sizes[0] / NFIELD;          // 32768
    const int wavesPerBlock = 8;                 // 256 threads = 8 wave32
    const int blocks = (B + wavesPerBlock - 1) / wavesPerBlock;
    ffm_kernel<<<blocks, 256, 0, stream>>>(inputs, w0, w, v, out, B);
}